// OneHotProjected_54881092108376
// MI455X (gfx1250) — compile-verified
//
#include <hip/hip_runtime.h>
#include <hip/hip_bf16.h>

#define VOCAB   17440
#define PROJ    256
#define PAD_IDX 0

typedef float v4f __attribute__((ext_vector_type(4)));

// ---------------------------------------------------------------------------
// Kernel 1: transpose W [PROJ, VOCAB] -> Wt [VOCAB, PROJ], 32x32 LDS tiles.
// VOCAB = 545*32, PROJ = 8*32 -> exact tiling, no bounds checks.
// blockDim (8,32): each thread moves one float4 in and one float4 out
// (global_load_b128 / global_store_b128, fully coalesced). LDS row stride 33
// keeps both the element-wise stores and the transposed reads conflict-free.
// ---------------------------------------------------------------------------
__global__ __launch_bounds__(256) void OneHot_transposeW_kernel(
    const float* __restrict__ W, float* __restrict__ Wt) {
    __shared__ float tile[32][33];
    const int vTile = blockIdx.x * 32;        // along VOCAB
    const int pTile = blockIdx.y * 32;        // along PROJ
    const int tx = threadIdx.x;               // 0..7  (float4 slot)
    const int ty = threadIdx.y;               // 0..31 (row)

    // Load: row p = pTile+ty, cols v = vTile + 4*tx .. +3  (b128, coalesced)
    const v4f in = *(const v4f*)&W[(size_t)(pTile + ty) * VOCAB
                                   + (size_t)(vTile + tx * 4)];
#pragma unroll
    for (int k = 0; k < 4; ++k)
        tile[ty][tx * 4 + k] = in[k];         // bank-conflict-free (stride 33)

    __syncthreads();

    // Store: row v = vTile+ty, cols p = pTile + 4*tx .. +3  (b128, coalesced)
    v4f outv;
#pragma unroll
    for (int k = 0; k < 4; ++k)
        outv[k] = tile[tx * 4 + k][ty];       // conflict-free reads
    *(v4f*)&Wt[(size_t)(vTile + ty) * PROJ + (size_t)(pTile + tx * 4)] = outv;
}

// ---------------------------------------------------------------------------
// Kernel 2: row gather from Wt with pad masking.
// 256 threads/block = 8 waves; 4 tokens/block; 64 threads per token, each
// moving one float4. Each wave32 covers half of one token's row, so the
// token index is wave-uniform: readfirstlane forces a scalar s_load of x and
// a scalar branch for the pad mask. Output stores are non-temporal
// (write-once stream) so the reusable 17.8 MB Wt stays resident in L2.
// ---------------------------------------------------------------------------
__global__ __launch_bounds__(256) void OneHot_gather_kernel(
    const int* __restrict__ x, const float* __restrict__ Wt,
    float* __restrict__ out, int n_tokens) {
    const int tid  = threadIdx.x;             // 0..255
    const int lane = tid & 63;                // float4 slot within 256-f row

    // Prefetch the Wt rows needed 4 blocks ahead (one 128B line per thread).
    // Emits gfx1250 global_prefetch_b8.
    if (tid < 32) {
        const int pf_token = (blockIdx.x + 4) * 4 + (tid >> 3);
        if (pf_token < n_tokens) {
            const int pf_idx = x[pf_token];
            if (pf_idx != PAD_IDX) {
                __builtin_prefetch(
                    Wt + (size_t)pf_idx * PROJ + (size_t)(tid & 7) * 32, 0, 3);
            }
        }
    }

    // token is uniform across each wave32 -> scalarize.
    const int token =
        __builtin_amdgcn_readfirstlane(blockIdx.x * 4 + (tid >> 6));
    if (token >= n_tokens) return;
    const int idx = __builtin_amdgcn_readfirstlane(x[token]); // s_load_b32

    v4f v = (v4f)(0.0f);
    if (idx != PAD_IDX) {                     // scalar branch, no divergence
        v = *(const v4f*)(Wt + (size_t)idx * PROJ + (size_t)lane * 4);
    }
    // Non-temporal b128 store: stream the 128 MB output past L2.
    __builtin_nontemporal_store(
        v, (v4f*)(out + (size_t)token * PROJ + (size_t)lane * 4));
}

// ---------------------------------------------------------------------------
// Fallback if the workspace can't hold Wt (17.8 MB): gather straight from W.
// Writes stay coalesced; W (17.8 MB) lives in the 192 MB L2 after first touch.
// ---------------------------------------------------------------------------
__global__ __launch_bounds__(256) void OneHot_gather_direct_kernel(
    const int* __restrict__ x, const float* __restrict__ W,
    float* __restrict__ out, int n_tokens) {
    const long long gid = (long long)blockIdx.x * 256 + threadIdx.x;
    const int token = (int)(gid >> 8);
    const int p     = (int)(gid & 255);
    if (token >= n_tokens) return;
    const int idx = x[token];
    const float val = (idx == PAD_IDX) ? 0.0f
                                       : W[(size_t)p * VOCAB + (size_t)idx];
    __builtin_nontemporal_store(val, &out[gid]);
}

extern "C" void kernel_launch(void* const* d_in, const int* in_sizes, int n_in,
                              void* d_out, int out_size, void* d_ws, size_t ws_size,
                              hipStream_t stream) {
    const int*   x   = (const int*)d_in[0];    // token ids, [64*2048]
    const float* W   = (const float*)d_in[1];  // [PROJ, VOCAB] f32
    float*       out = (float*)d_out;          // [n_tokens, PROJ] f32
    const int n_tokens = in_sizes[0];

    const size_t wt_bytes = (size_t)VOCAB * PROJ * sizeof(float);
    if (ws_size >= wt_bytes) {
        float* Wt = (float*)d_ws;
        dim3 tgrid(VOCAB / 32, PROJ / 32);     // 545 x 8 tiles
        dim3 tblock(8, 32);
        OneHot_transposeW_kernel<<<tgrid, tblock, 0, stream>>>(W, Wt);

        const int nblocks = (n_tokens + 3) / 4;
        OneHot_gather_kernel<<<nblocks, 256, 0, stream>>>(x, Wt, out, n_tokens);
    } else {
        const long long total = (long long)n_tokens * PROJ;
        const int nblocks = (int)((total + 255) / 256);
        OneHot_gather_direct_kernel<<<nblocks, 256, 0, stream>>>(x, W, out, n_tokens);
    }
}